// E3Conv_77455440216501
// MI455X (gfx1250) — compile-verified
//
#include <hip/hip_runtime.h>
#include <math.h>

typedef __attribute__((ext_vector_type(2))) float v2f;
typedef __attribute__((ext_vector_type(8))) float v8f;

#define HIDC 32
#define KSEG 288      // 9 (hh) * 32 (channels)
#define KP   292      // padded K stride (keeps b64 alignment, kills bank aliasing)

#define SQ3f      1.7320508075688772f
#define ISQ3f     0.5773502691896258f
#define C_INITf   0.17677669529663687f   // 1/sqrt(32)
#define C_HIDf    0.125f                 // 1/sqrt(64)
#define C_ETf     0.08838834764831843f   // 1/sqrt(128)
#define LCf       0.17677669529663687f   // 1/sqrt(32)
#define ISQ13f    0.2773500981126146f    // 1/sqrt(13)

__device__ __forceinline__ float silu_f(float x) { return x / (1.0f + __expf(-x)); }
__device__ __forceinline__ void atomAddF(float* p, float v) { unsafeAtomicAdd(p, v); }

// ---------------------------------------------------------------------------
// Core fused-TP GEMM segment: acc[t] += (hh*scale ⊗ x) @ Bt  over K=288.
// A-operand (16x4 f32):  lane m=lane&15 is row (edge), kofs = lane<16 ? {0,1} : {2,3}
// B-operand (4x16 f32):  lane col = lane&15, same K split; Bt stored [col][K(pad)]
// D (16x16 f32, v8f):    lane col = lane&15, rows M = r + 8*(lane>=16)
// ---------------------------------------------------------------------------
template<int NCT>
__device__ __forceinline__ void gemm288(const float* Bt, int cstride,
                                        const float* xblk,       // LDS [16][32]
                                        const float* hh,         // 9 per-lane regs
                                        float scale, int m, int kofs, int colq,
                                        v8f* acc)
{
    v2f sp[8];
#pragma unroll
    for (int q = 0; q < 8; ++q)
        sp[q] = *(const v2f*)(xblk + m * HIDC + q * 4 + kofs);
#pragma unroll
    for (int c8 = 0; c8 < 9; ++c8) {
        float hv = hh[c8] * scale;
#pragma unroll
        for (int q = 0; q < 8; ++q) {
            int kl = c8 * 32 + q * 4 + kofs;
            v2f a = sp[q] * hv;
#pragma unroll
            for (int t = 0; t < NCT; ++t) {
                v2f b = *(const v2f*)(Bt + (t * 16 + colq) * cstride + kl);
                acc[t] = __builtin_amdgcn_wmma_f32_16x16x4_f32(
                    false, a, false, b, (short)0, acc[t], false, false);
            }
        }
    }
}

// ---------------------------------------------------------------------------
// Kernel 1: per-edge SH, gaussian radial basis, three MLP hiddens, degree cnt
// ---------------------------------------------------------------------------
__global__ void k_edge_pre(const float* __restrict__ coord, const int* __restrict__ EI,
                           const float* ipW0, const float* ipb0,
                           const float* lyW0, const float* lyb0,
                           const float* etW0, const float* etb0,
                           float* sh1g, float* hipg, float* hlyg, float* hetg,
                           float* cnt, int E)
{
    int e = blockIdx.x * blockDim.x + threadIdx.x;
    if (e >= E) return;
    int sn = EI[e], dn = EI[E + e];
    float vx = coord[dn * 3 + 0] - coord[sn * 3 + 0];
    float vy = coord[dn * 3 + 1] - coord[sn * 3 + 1];
    float vz = coord[dn * 3 + 2] - coord[sn * 3 + 2];
    float d = sqrtf(vx * vx + vy * vy + vz * vz + 1e-12f);
    float inv = 1.0f / d;
    sh1g[e * 3 + 0] = SQ3f * vy * inv;   // e3nn (y,z,x)
    sh1g[e * 3 + 1] = SQ3f * vz * inv;
    sh1g[e * 3 + 2] = SQ3f * vx * inv;
    float attr[8];
    const float step = 5.0f / 9.0f;
#pragma unroll
    for (int i = 0; i < 8; ++i) {
        float df = (d - step * (float)(i + 1)) / step;
        attr[i] = __expf(-df * df) * (1.0f / 1.12f);
    }
#pragma unroll
    for (int k = 0; k < 8; ++k) {
        float a0 = ipb0[k], a1 = lyb0[k], a2 = etb0[k];
#pragma unroll
        for (int j = 0; j < 8; ++j) {
            float aj = attr[j];
            a0 += aj * ipW0[j * 8 + k];
            a1 += aj * lyW0[j * 8 + k];
            a2 += aj * etW0[j * 8 + k];
        }
        hipg[e * 8 + k] = silu_f(a0);
        hlyg[e * 8 + k] = silu_f(a1);
        hetg[e * 8 + k] = silu_f(a2);
    }
    atomAddF(&cnt[dn], 1.0f);
}

// ---------------------------------------------------------------------------
// Weight folding: pack W1/b1 into [col][Kpad] B-operand layout (bias = hh[8])
// ---------------------------------------------------------------------------
__global__ void k_fold_ip(const float* __restrict__ W1, const float* __restrict__ b1,
                          float* __restrict__ out)
{
    int idx = blockIdx.x * blockDim.x + threadIdx.x;
    if (idx >= 64 * KP) return;
    int c = idx / KP, k = idx % KP;
    float v = 0.0f;
    if (k < KSEG) {
        int kb = k >> 5, u = k & 31;
        int mcol = (c < 32) ? (u * 32 + c) : (1024 + u * 32 + (c - 32));
        v = (kb < 8) ? W1[kb * 2048 + mcol] : b1[mcol];
    }
    out[idx] = v;
}

__global__ void k_fold_ly(const float* __restrict__ W1, const float* __restrict__ b1,
                          float* __restrict__ out)
{
    int idx = blockIdx.x * blockDim.x + threadIdx.x;
    if (idx >= 4 * 32 * KP) return;
    int j = idx / (32 * KP), rem = idx % (32 * KP);
    int c = rem / KP, k = rem % KP;
    float v = 0.0f;
    if (k < KSEG) {
        int kb = k >> 5, u = k & 31;
        int mcol = j * 1024 + u * 32 + c;
        v = (kb < 8) ? W1[kb * 4096 + mcol] : b1[mcol];
    }
    out[idx] = v;
}

__global__ void k_fold_et(const float* __restrict__ W1, const float* __restrict__ b1,
                          float* __restrict__ out)
{
    int idx = blockIdx.x * blockDim.x + threadIdx.x;
    if (idx >= 16 * (4 * KP)) return;
    int c = idx / (4 * KP), kt = idx % (4 * KP);
    int j = kt / KP, kl = kt % KP;
    float v = 0.0f;
    if (kl < KSEG && c < 5) {
        int kb = kl >> 5, u = kl & 31;
        int mcol = j * 160 + u * 5 + c;
        v = (kb < 8) ? W1[kb * 640 + mcol] : b1[mcol];
    }
    out[idx] = v;
}

// ---------------------------------------------------------------------------
// Kernel: initial projector  Conv(32x0e -> 32x0e + 32x1o), WMMA + atomic scatter
// ---------------------------------------------------------------------------
__global__ void k_tp_init(const int* __restrict__ EI, const int* __restrict__ atype,
                          const float* __restrict__ embed,
                          const float* __restrict__ hipg, const float* __restrict__ sh1g,
                          const float* __restrict__ Wip,
                          float* s1a, float* v1a, int E, int ntiles)
{
    extern __shared__ float sm[];
    const int BN = 64 * KP;
    for (int i = threadIdx.x; i < BN; i += blockDim.x) sm[i] = Wip[i];
    __syncthreads();
    int wave = threadIdx.x >> 5, lane = threadIdx.x & 31;
    int tile = blockIdx.x * (blockDim.x >> 5) + wave;
    float* sw = sm + BN + wave * 512;
    int base = tile * 16;
    if (tile < ntiles) {
        for (int idx = lane; idx < 512; idx += 32) {
            int ml = idx >> 5, u = idx & 31, e = base + ml;
            float val = 0.0f;
            if (e < E) { int sn = EI[e]; val = embed[atype[sn] * HIDC + u]; }
            sw[ml * HIDC + u] = val;
        }
    }
    __syncthreads();
    if (tile >= ntiles) return;
    int m = lane & 15, kofs = (lane >= 16) ? 2 : 0, colq = m;
    int eh = min(base + m, E - 1);
    float hh[9];
#pragma unroll
    for (int k = 0; k < 8; ++k) hh[k] = hipg[eh * 8 + k];
    hh[8] = 1.0f;
    v8f acc[4];
#pragma unroll
    for (int t = 0; t < 4; ++t) acc[t] = (v8f){0, 0, 0, 0, 0, 0, 0, 0};
    gemm288<4>(sm, KP, sw, hh, 1.0f, m, kofs, colq, acc);
    for (int r = 0; r < 8; ++r) {
        int e = base + r + ((lane >= 16) ? 8 : 0);
        if (e >= E) continue;
        int dn = EI[E + e];
        float sx = sh1g[e * 3 + 0], sy = sh1g[e * 3 + 1], sz = sh1g[e * 3 + 2];
        atomAddF(&s1a[dn * HIDC + colq],      C_INITf * acc[0][r]);
        atomAddF(&s1a[dn * HIDC + colq + 16], C_INITf * acc[1][r]);
        float t0 = C_INITf * acc[2][r], t1 = C_INITf * acc[3][r];
        atomAddF(&v1a[dn * 96 + colq * 3 + 0], t0 * sx);
        atomAddF(&v1a[dn * 96 + colq * 3 + 1], t0 * sy);
        atomAddF(&v1a[dn * 96 + colq * 3 + 2], t0 * sz);
        atomAddF(&v1a[dn * 96 + (colq + 16) * 3 + 0], t1 * sx);
        atomAddF(&v1a[dn * 96 + (colq + 16) * 3 + 1], t1 * sy);
        atomAddF(&v1a[dn * 96 + (colq + 16) * 3 + 2], t1 * sz);
    }
}

// ---------------------------------------------------------------------------
// scatter-mean normalize (after init conv)
// ---------------------------------------------------------------------------
__global__ void k_fin1(const float* cnt, const float* s1a, const float* v1a,
                       float* sbuf, float* vbuf, int N)
{
    int idx = blockIdx.x * blockDim.x + threadIdx.x;
    if (idx >= N * 128) return;
    int n = idx >> 7, r = idx & 127;
    float ci = 1.0f / fmaxf(cnt[n], 1.0f);
    if (r < 32) sbuf[n * 32 + r] = s1a[n * 32 + r] * ci;
    else        vbuf[n * 96 + (r - 32)] = v1a[n * 96 + (r - 32)] * ci;
}

// ---------------------------------------------------------------------------
// Kernel: hidden conv  FCTP(32x0e+32x1o, 1x0e+1x1o -> 32x0e+32x1o)
// out_s = c(z_s@W0 + z_dot@W3);  out_v_i = c(z_{s*sh1_i}@W1 + z_{v_i}@W2)
// ---------------------------------------------------------------------------
__global__ void k_tp_hidden(const int* __restrict__ EI,
                            const float* __restrict__ sbuf, const float* __restrict__ vbuf,
                            const float* __restrict__ hlyg, const float* __restrict__ sh1g,
                            const float* __restrict__ Wly,
                            float* s2a, float* v2a, int E, int ntiles)
{
    extern __shared__ float sm[];
    const int BN = 4 * 32 * KP;
    for (int i = threadIdx.x; i < BN; i += blockDim.x) sm[i] = Wly[i];
    __syncthreads();
    int wave = threadIdx.x >> 5, lane = threadIdx.x & 31;
    int tile = blockIdx.x * (blockDim.x >> 5) + wave;
    float* sw = sm + BN + wave * 2560;   // s[512] | dot[512] | v0 | v1 | v2
    int base = tile * 16;
    if (tile < ntiles) {
        for (int idx = lane; idx < 512; idx += 32) {
            int ml = idx >> 5, u = idx & 31, e = base + ml;
            float sv = 0, dv = 0, w0 = 0, w1 = 0, w2 = 0;
            if (e < E) {
                int sn = EI[e];
                sv = sbuf[sn * 32 + u];
                w0 = vbuf[sn * 96 + u * 3 + 0];
                w1 = vbuf[sn * 96 + u * 3 + 1];
                w2 = vbuf[sn * 96 + u * 3 + 2];
                dv = (w0 * sh1g[e * 3 + 0] + w1 * sh1g[e * 3 + 1] + w2 * sh1g[e * 3 + 2]) * ISQ3f;
            }
            sw[ml * 32 + u]        = sv;
            sw[512 + ml * 32 + u]  = dv;
            sw[1024 + ml * 32 + u] = w0;
            sw[1536 + ml * 32 + u] = w1;
            sw[2048 + ml * 32 + u] = w2;
        }
    }
    __syncthreads();
    if (tile >= ntiles) return;
    int m = lane & 15, kofs = (lane >= 16) ? 2 : 0, colq = m;
    int eh = min(base + m, E - 1);
    float hh[9];
#pragma unroll
    for (int k = 0; k < 8; ++k) hh[k] = hlyg[eh * 8 + k];
    hh[8] = 1.0f;
    float sm0 = sh1g[eh * 3 + 0], sm1 = sh1g[eh * 3 + 1], sm2 = sh1g[eh * 3 + 2];
    v8f aS[2], aV0[2], aV1[2], aV2[2];
#pragma unroll
    for (int t = 0; t < 2; ++t) {
        aS[t] = (v8f){0,0,0,0,0,0,0,0}; aV0[t] = (v8f){0,0,0,0,0,0,0,0};
        aV1[t] = (v8f){0,0,0,0,0,0,0,0}; aV2[t] = (v8f){0,0,0,0,0,0,0,0};
    }
    gemm288<2>(sm + 0 * 32 * KP, KP, sw,        hh, 1.0f, m, kofs, colq, aS);
    gemm288<2>(sm + 3 * 32 * KP, KP, sw + 512,  hh, 1.0f, m, kofs, colq, aS);
    gemm288<2>(sm + 1 * 32 * KP, KP, sw,        hh, sm0,  m, kofs, colq, aV0);
    gemm288<2>(sm + 2 * 32 * KP, KP, sw + 1024, hh, 1.0f, m, kofs, colq, aV0);
    gemm288<2>(sm + 1 * 32 * KP, KP, sw,        hh, sm1,  m, kofs, colq, aV1);
    gemm288<2>(sm + 2 * 32 * KP, KP, sw + 1536, hh, 1.0f, m, kofs, colq, aV1);
    gemm288<2>(sm + 1 * 32 * KP, KP, sw,        hh, sm2,  m, kofs, colq, aV2);
    gemm288<2>(sm + 2 * 32 * KP, KP, sw + 2048, hh, 1.0f, m, kofs, colq, aV2);
    for (int r = 0; r < 8; ++r) {
        int e = base + r + ((lane >= 16) ? 8 : 0);
        if (e >= E) continue;
        int dn = EI[E + e];
        atomAddF(&s2a[dn * 32 + colq],      C_HIDf * aS[0][r]);
        atomAddF(&s2a[dn * 32 + colq + 16], C_HIDf * aS[1][r]);
        atomAddF(&v2a[dn * 96 + colq * 3 + 0],        C_HIDf * aV0[0][r]);
        atomAddF(&v2a[dn * 96 + (colq + 16) * 3 + 0], C_HIDf * aV0[1][r]);
        atomAddF(&v2a[dn * 96 + colq * 3 + 1],        C_HIDf * aV1[0][r]);
        atomAddF(&v2a[dn * 96 + (colq + 16) * 3 + 1], C_HIDf * aV1[1][r]);
        atomAddF(&v2a[dn * 96 + colq * 3 + 2],        C_HIDf * aV2[0][r]);
        atomAddF(&v2a[dn * 96 + (colq + 16) * 3 + 2], C_HIDf * aV2[1][r]);
    }
}

// ---------------------------------------------------------------------------
// scatter-mean + o3.Linear self-interaction:  s = cs + lc*s@Ws ; v likewise
// ---------------------------------------------------------------------------
__global__ void k_fin2(const float* cnt, const float* s2a, const float* v2a,
                       const float* sbuf, const float* vbuf,
                       const float* lyWs, const float* lyWv,
                       float* sfin, float* vfin, int N)
{
    int idx = blockIdx.x * blockDim.x + threadIdx.x;
    if (idx >= N * 32) return;
    int n = idx >> 5, w = idx & 31;
    float ci = 1.0f / fmaxf(cnt[n], 1.0f);
    float s  = s2a[n * 32 + w] * ci;
    float a0 = v2a[n * 96 + w * 3 + 0] * ci;
    float a1 = v2a[n * 96 + w * 3 + 1] * ci;
    float a2 = v2a[n * 96 + w * 3 + 2] * ci;
    for (int u = 0; u < 32; ++u) {
        float Ws = LCf * lyWs[u * 32 + w], Wv = LCf * lyWv[u * 32 + w];
        s  += sbuf[n * 32 + u] * Ws;
        a0 += vbuf[n * 96 + u * 3 + 0] * Wv;
        a1 += vbuf[n * 96 + u * 3 + 1] * Wv;
        a2 += vbuf[n * 96 + u * 3 + 2] * Wv;
    }
    sfin[n * 32 + w] = s;
    vfin[n * 96 + w * 3 + 0] = a0;
    vfin[n * 96 + w * 3 + 1] = a1;
    vfin[n * 96 + w * 3 + 2] = a2;
}

// ---------------------------------------------------------------------------
// Kernel: edge output FCTP([h_src,h_dst], sh -> 5x0e), 4x288-K WMMA, N padded to 16
// ---------------------------------------------------------------------------
__global__ void k_edge_out(const int* __restrict__ EI,
                           const float* __restrict__ sfin, const float* __restrict__ vfin,
                           const float* __restrict__ hetg, const float* __restrict__ sh1g,
                           const float* __restrict__ Wet,
                           float* __restrict__ outE, int E, int ntiles)
{
    extern __shared__ float sm[];
    const int BN = 16 * (4 * KP);
    for (int i = threadIdx.x; i < BN; i += blockDim.x) sm[i] = Wet[i];
    __syncthreads();
    int wave = threadIdx.x >> 5, lane = threadIdx.x & 31;
    int tile = blockIdx.x * (blockDim.x >> 5) + wave;
    float* sw = sm + BN + wave * 2048;   // s1 | d1 | s2 | d2, each [16][32]
    int base = tile * 16;
    if (tile < ntiles) {
        for (int idx = lane; idx < 512; idx += 32) {
            int ml = idx >> 5, u = idx & 31, e = base + ml;
            float s1 = 0, d1 = 0, s2 = 0, d2 = 0;
            if (e < E) {
                int sn = EI[e], dn = EI[E + e];
                float x0 = sh1g[e * 3 + 0], x1 = sh1g[e * 3 + 1], x2 = sh1g[e * 3 + 2];
                s1 = sfin[sn * 32 + u];
                s2 = sfin[dn * 32 + u];
                d1 = (vfin[sn * 96 + u * 3 + 0] * x0 + vfin[sn * 96 + u * 3 + 1] * x1 +
                      vfin[sn * 96 + u * 3 + 2] * x2) * ISQ3f;
                d2 = (vfin[dn * 96 + u * 3 + 0] * x0 + vfin[dn * 96 + u * 3 + 1] * x1 +
                      vfin[dn * 96 + u * 3 + 2] * x2) * ISQ3f;
            }
            sw[ml * 32 + u]        = s1;
            sw[512 + ml * 32 + u]  = d1;
            sw[1024 + ml * 32 + u] = s2;
            sw[1536 + ml * 32 + u] = d2;
        }
    }
    __syncthreads();
    if (tile >= ntiles) return;
    int m = lane & 15, kofs = (lane >= 16) ? 2 : 0, colq = m;
    int eh = min(base + m, E - 1);
    float hh[9];
#pragma unroll
    for (int k = 0; k < 8; ++k) hh[k] = hetg[eh * 8 + k];
    hh[8] = 1.0f;
    v8f acc[1] = {(v8f){0,0,0,0,0,0,0,0}};
#pragma unroll
    for (int j = 0; j < 4; ++j)
        gemm288<1>(sm + j * KP, 4 * KP, sw + j * 512, hh, 1.0f, m, kofs, colq, acc);
    if (colq < 5) {
        for (int r = 0; r < 8; ++r) {
            int e = base + r + ((lane >= 16) ? 8 : 0);
            if (e < E) outE[e * 5 + colq] = C_ETf * acc[0][r];
        }
    }
}

// ---------------------------------------------------------------------------
// node output MLP (13x0e)
// ---------------------------------------------------------------------------
__global__ void k_node_out(const float* __restrict__ sfin, const float* __restrict__ W1,
                           const float* __restrict__ W2, float* __restrict__ outN, int N)
{
    int n = blockIdx.x * blockDim.x + threadIdx.x;
    if (n >= N) return;
    float h[13];
#pragma unroll
    for (int j = 0; j < 13; ++j) {
        float a = 0.0f;
        for (int u = 0; u < 32; ++u) a += sfin[n * 32 + u] * W1[u * 13 + j];
        h[j] = silu_f(a * LCf);
    }
#pragma unroll
    for (int o = 0; o < 13; ++o) {
        float a = 0.0f;
#pragma unroll
        for (int j = 0; j < 13; ++j) a += h[j] * W2[j * 13 + o];
        outN[n * 13 + o] = a * ISQ13f;
    }
}

// ---------------------------------------------------------------------------
extern "C" void kernel_launch(void* const* d_in, const int* in_sizes, int n_in,
                              void* d_out, int out_size, void* d_ws, size_t ws_size,
                              hipStream_t stream)
{
    const float* coord = (const float*)d_in[0];
    const int*   atype = (const int*)d_in[1];
    const int*   EI    = (const int*)d_in[2];
    const float* embed = (const float*)d_in[3];
    const float* ipW0 = (const float*)d_in[4];  const float* ipb0 = (const float*)d_in[5];
    const float* ipW1 = (const float*)d_in[6];  const float* ipb1 = (const float*)d_in[7];
    const float* lyW0 = (const float*)d_in[8];  const float* lyb0 = (const float*)d_in[9];
    const float* lyW1 = (const float*)d_in[10]; const float* lyb1 = (const float*)d_in[11];
    const float* lyWs = (const float*)d_in[12]; const float* lyWv = (const float*)d_in[13];
    const float* etW0 = (const float*)d_in[14]; const float* etb0 = (const float*)d_in[15];
    const float* etW1 = (const float*)d_in[16]; const float* etb1 = (const float*)d_in[17];
    const float* noW1 = (const float*)d_in[18]; const float* noW2 = (const float*)d_in[19];

    const int N = in_sizes[0] / 3;
    const int E = in_sizes[2] / 2;
    const int ntiles = (E + 15) / 16;

    float* outE = (float*)d_out;
    float* outN = outE + (size_t)E * 5;

    // workspace layout (floats)
    float* ws = (float*)d_ws;
    size_t o = 0;
    float* SH1 = ws + o; o += (size_t)E * 3;
    float* HIP = ws + o; o += (size_t)E * 8;
    float* HLY = ws + o; o += (size_t)E * 8;
    float* HET = ws + o; o += (size_t)E * 8;
    float* CNT = ws + o; size_t accBeg = o; o += N;            // --- accumulators
    float* S1A = ws + o; o += (size_t)N * 32;
    float* V1A = ws + o; o += (size_t)N * 96;
    float* S2A = ws + o; o += (size_t)N * 32;
    float* V2A = ws + o; size_t accEnd = o + (size_t)N * 96; o = accEnd;
    float* SB  = ws + o; o += (size_t)N * 32;
    float* VB  = ws + o; o += (size_t)N * 96;
    float* SF  = ws + o; o += (size_t)N * 32;
    float* VF  = ws + o; o += (size_t)N * 96;
    float* WIP = ws + o; o += (size_t)64 * KP;
    float* WLY = ws + o; o += (size_t)4 * 32 * KP;
    float* WET = ws + o; o += (size_t)16 * (4 * KP);

    // zero the atomic accumulators (contiguous region)
    hipMemsetAsync(ws + accBeg, 0, (accEnd - accBeg) * sizeof(float), stream);

    // 1) per-edge preprocessing
    k_edge_pre<<<(E + 255) / 256, 256, 0, stream>>>(coord, EI, ipW0, ipb0, lyW0, lyb0,
                                                    etW0, etb0, SH1, HIP, HLY, HET, CNT, E);
    // 2) fold weights into WMMA-B layout
    k_fold_ip<<<(64 * KP + 255) / 256, 256, 0, stream>>>(ipW1, ipb1, WIP);
    k_fold_ly<<<(4 * 32 * KP + 255) / 256, 256, 0, stream>>>(lyW1, lyb1, WLY);
    k_fold_et<<<(16 * 4 * KP + 255) / 256, 256, 0, stream>>>(etW1, etb1, WET);

    // 3) initial projector conv (8 waves/block share B in LDS)
    {
        int wpb = 8, threads = wpb * 32;
        int blocks = (ntiles + wpb - 1) / wpb;
        size_t shmem = (size_t)(64 * KP + wpb * 512) * sizeof(float);   // ~91 KB
        k_tp_init<<<blocks, threads, shmem, stream>>>(EI, atype, embed, HIP, SH1, WIP,
                                                      S1A, V1A, E, ntiles);
    }
    k_fin1<<<(N * 128 + 255) / 256, 256, 0, stream>>>(CNT, S1A, V1A, SB, VB, N);

    // 4) hidden conv (4 waves/block, ~186 KB LDS)
    {
        int wpb = 4, threads = wpb * 32;
        int blocks = (ntiles + wpb - 1) / wpb;
        size_t shmem = (size_t)(4 * 32 * KP + wpb * 2560) * sizeof(float);
        k_tp_hidden<<<blocks, threads, shmem, stream>>>(EI, SB, VB, HLY, SH1, WLY,
                                                        S2A, V2A, E, ntiles);
    }
    k_fin2<<<(N * 32 + 255) / 256, 256, 0, stream>>>(CNT, S2A, V2A, SB, VB, lyWs, lyWv,
                                                     SF, VF, N);

    // 5) edge output TP (4 waves/block, ~105 KB LDS)
    {
        int wpb = 4, threads = wpb * 32;
        int blocks = (ntiles + wpb - 1) / wpb;
        size_t shmem = (size_t)(16 * 4 * KP + wpb * 2048) * sizeof(float);
        k_edge_out<<<blocks, threads, shmem, stream>>>(EI, SF, VF, HET, SH1, WET,
                                                       outE, E, ntiles);
    }
    // 6) node output MLP
    k_node_out<<<(N + 255) / 256, 256, 0, stream>>>(SF, noW1, noW2, outN, N);
}